// DenseDilatedKnnGraph_20289425506351
// MI455X (gfx1250) — compile-verified
//
#include <hip/hip_runtime.h>
#include <hip/hip_bf16.h>
#include <stdint.h>

// Problem constants (from reference): x (4, 64, 8192, 1), K=16, DILATION=2
#define BN   4
#define NN   8192
#define DD   64
#define KOUT 16
#define KSEL 32   // K * DILATION, size of top-k kept before dilation stride

typedef float v2f __attribute__((ext_vector_type(2)));
typedef float v8f __attribute__((ext_vector_type(8)));

__device__ __forceinline__ unsigned lane_id() {
  return __builtin_amdgcn_mbcnt_lo(~0u, 0u);   // wave32: lo only
}

__device__ __forceinline__ int bperm32(int src_lane, int v) {
  return __builtin_amdgcn_ds_bpermute(src_lane << 2, v);
}

__device__ __forceinline__ uint64_t bperm64(int src_lane, uint64_t v) {
  int lo = bperm32(src_lane, (int)(uint32_t)v);
  int hi = bperm32(src_lane, (int)(uint32_t)(v >> 32));
  return ((uint64_t)(uint32_t)hi << 32) | (uint32_t)lo;
}

__device__ __forceinline__ uint64_t shflx64(uint64_t v, unsigned xm) {
  unsigned l = lane_id();
  return bperm64((int)((l ^ xm) & 31u), v);
}

// Monotone order-preserving map float -> uint32 (larger float => larger uint)
__device__ __forceinline__ uint32_t ford(float f) {
  uint32_t u = __float_as_uint(f);
  return (u & 0x80000000u) ? ~u : (u | 0x80000000u);
}

// ---------------- Kernel 1: squared norms per point ----------------
__global__ void norms_kernel(const float* __restrict__ x, float* __restrict__ norms) {
  int t = blockIdx.x * blockDim.x + threadIdx.x;      // 0 .. B*N-1
  int b = t >> 13;                                     // t / 8192
  int n = t & (NN - 1);
  const float* xb = x + (size_t)b * DD * NN + n;
  float s = 0.0f;
  #pragma unroll
  for (int d = 0; d < DD; ++d) {
    float v = xb[(size_t)d * NN];
    s += v * v;
  }
  norms[t] = s;
}

// ---------------- Kernel 2: fused WMMA distance + top-32 select ----------------
// Block: 256 threads (8 waves). One block per (batch, 16-row tile).
__global__ void __launch_bounds__(256)
knn_kernel(const float* __restrict__ x, const float* __restrict__ norms,
           int* __restrict__ out) {
  __shared__ uint64_t lists[8][16][32];   // per-wave, per-row partial top-32 (32 KB)
  __shared__ uint64_t mins[8][16];        // per-wave, per-row current list minimum
  __shared__ uint32_t mins_hi[8][16];     // high word (value part) of mins, fast filter
  __shared__ float    rown[16];           // row norms for this tile

  const int b   = blockIdx.x / (NN / 16);
  const int tm  = blockIdx.x % (NN / 16);
  const int m0  = tm * 16;

  const unsigned wv    = threadIdx.x >> 5;
  const unsigned lane  = threadIdx.x & 31u;
  const unsigned lh    = lane & 15u;     // position within lane-half
  const unsigned khalf = lane >> 4;      // which K pair (A/B f32 16x16x4 layout)

  // --- init LDS ---
  {
    uint64_t* lp = &lists[0][0][0];
    for (int i = threadIdx.x; i < 8 * 16 * 32; i += 256) lp[i] = 0ull;
    if (threadIdx.x < 128) {
      (&mins[0][0])[threadIdx.x]    = 0ull;
      (&mins_hi[0][0])[threadIdx.x] = 0u;
    }
    if (threadIdx.x < 16)  rown[threadIdx.x] = norms[(size_t)b * NN + m0 + threadIdx.x];
  }
  __syncthreads();

  const float* xb = x + (size_t)b * DD * NN;

  // --- preload A tile (16 rows x 64 dims) in registers: 16 k-steps, v2f each ---
  const int arow = m0 + (int)lh;
  v2f A[16];
  #pragma unroll
  for (int kk = 0; kk < 16; ++kk) {
    int d0 = kk * 4 + (int)khalf * 2;
    A[kk].x = xb[(size_t)d0 * NN + arow];
    A[kk].y = xb[(size_t)(d0 + 1) * NN + arow];
  }

  float nrow[8];
  #pragma unroll
  for (int r = 0; r < 8; ++r) nrow[r] = rown[r + (int)khalf * 8];

  // --- main loop: wave wv handles column tiles wv, wv+8, ... ---
  for (int j = (int)wv; j < NN / 16; j += 8) {
    const int col = j * 16 + (int)lh;

    // speculative prefetch of next strip for this wave (dropped if OOB)
    __builtin_prefetch(&xb[col + 128], 0, 0);

    v8f C = {};
    #pragma unroll
    for (int kk = 0; kk < 16; ++kk) {
      int d0 = kk * 4 + (int)khalf * 2;
      v2f Bv;
      Bv.x = xb[(size_t)d0 * NN + col];
      Bv.y = xb[(size_t)(d0 + 1) * NN + col];
      C = __builtin_amdgcn_wmma_f32_16x16x4_f32(
              /*neg_a=*/false, A[kk], /*neg_b=*/false, Bv,
              /*c_mod=*/(short)0, C, /*reuse_a=*/false, /*reuse_b=*/false);
    }

    const float ncol = norms[(size_t)b * NN + col];

    #pragma unroll
    for (int r = 0; r < 8; ++r) {
      const int rowc = r + (int)khalf * 8;            // C layout: VGPR r -> row r / r+8
      const float v  = 2.0f * C[r] - nrow[r] - ncol;  // neg_adj (bigger = closer)
      const uint32_t khi = ford(v);
      const uint64_t key = ((uint64_t)khi << 32) | (uint32_t)(~(uint32_t)col);

      // Fast filter: 32-bit compare vs value-part of current min (conservative
      // on ties; insert path does the exact 64-bit recheck).
      unsigned mask = __builtin_amdgcn_ballot_w32(khi >= mins_hi[wv][rowc]);

      while (mask) {
        const int src = __ffs(mask) - 1;
        mask &= mask - 1;
        const uint64_t k  = bperm64(src, key);
        const int      rc = bperm32(src, rowc);

        // cooperative replace-min in lists[wv][rc] (one entry per lane)
        const uint64_t e = lists[wv][rc][lane];
        uint64_t mv = e; unsigned ml = lane;
        #pragma unroll
        for (int off = 16; off; off >>= 1) {
          uint64_t ov = shflx64(mv, off);
          unsigned ol = (unsigned)bperm32((int)((lane ^ off) & 31u), (int)ml);
          if (ov < mv || (ov == mv && ol < ml)) { mv = ov; ml = ol; }
        }
        if (k > mv) {
          if (lane == ml) lists[wv][rc][lane] = k;
          uint64_t e2 = (lane == ml) ? k : e;
          uint64_t m2 = e2;
          #pragma unroll
          for (int off = 16; off; off >>= 1) {
            uint64_t ov = shflx64(m2, off);
            if (ov < m2) m2 = ov;
          }
          if (lane == 0) {
            mins[wv][rc]    = m2;
            mins_hi[wv][rc] = (uint32_t)(m2 >> 32);
          }
        }
      }
    }
  }

  __syncthreads();

  // --- merge: wave wv handles rows 2*wv and 2*wv+1 ---
  #pragma unroll
  for (int rr = 0; rr < 2; ++rr) {
    const int rowc = (int)wv * 2 + rr;
    const int grow = m0 + rowc;

    uint64_t cand[8];
    #pragma unroll
    for (int w2 = 0; w2 < 8; ++w2) cand[w2] = lists[w2][rowc][lane];

    for (int rank = 0; rank < KSEL; ++rank) {
      uint64_t lm = 0; int lw = -1;
      #pragma unroll
      for (int w2 = 0; w2 < 8; ++w2)
        if (cand[w2] > lm) { lm = cand[w2]; lw = w2; }

      uint64_t gm = lm;
      #pragma unroll
      for (int off = 16; off; off >>= 1) {
        uint64_t ov = shflx64(gm, off);
        if (ov > gm) gm = ov;
      }
      if (lw >= 0 && lm == gm) cand[lw] = 0;   // keys unique -> one owner

      if ((rank & 1) == 0 && lane == 0) {      // dilation stride 2
        const int kk2    = rank >> 1;
        const int nn_idx = (int)(~(uint32_t)gm);
        const size_t base = ((size_t)b * NN + grow) * KOUT + kk2;
        out[base]                               = nn_idx;  // edge_index[0]
        out[(size_t)BN * NN * KOUT + base]      = grow;    // edge_index[1]
      }
    }
  }
}

extern "C" void kernel_launch(void* const* d_in, const int* in_sizes, int n_in,
                              void* d_out, int out_size, void* d_ws, size_t ws_size,
                              hipStream_t stream) {
  const float* x = (const float*)d_in[0];
  float* norms   = (float*)d_ws;          // needs B*N*4 = 128 KB
  int*   out     = (int*)d_out;

  norms_kernel<<<(BN * NN) / 256, 256, 0, stream>>>(x, norms);
  knn_kernel<<<BN * (NN / 16), 256, 0, stream>>>(x, norms, out);
}